// DecoderC_50225347560165
// MI455X (gfx1250) — compile-verified
//
#include <hip/hip_runtime.h>
#include <hip/hip_bf16.h>
#include <math.h>

// ---------------------------------------------------------------------------
// Sizes (from reference)
#define HID   512
#define EMB   128
#define CTX   512
#define VOCAB 256
#define BSZ   64
#define SEQ   4096
#define D1    1024
#define G3    (3*HID)   // 1536

typedef float v2f __attribute__((ext_vector_type(2)));
typedef float v8f __attribute__((ext_vector_type(8)));
typedef __attribute__((address_space(3))) float lds_f32;

// ---------------------------------------------------------------------------
// Workspace layout (float offsets)
#define OFF_SCORES 0                         // B*S            = 262144
#define OFF_SBIAS  (OFF_SCORES + BSZ*SEQ)    // 64
#define OFF_Y      (OFF_SBIAS + BSZ)         // B*EMB          = 8192
#define OFF_CPART  (OFF_Y + BSZ*EMB)         // B*8*CTX        = 262144
#define OFF_CTX    (OFF_CPART + BSZ*8*CTX)   // B*CTX          = 32768
#define OFF_GX     (OFF_CTX + BSZ*CTX)       // 2*B*G3         = 196608
#define OFF_GH     (OFF_GX + 2*BSZ*G3)       // 2*B*G3         = 196608
#define OFF_T      (OFF_GH + 2*BSZ*G3)       // B*D1           = 65536

// Output layout (float offsets): next_char | h | alpha
#define OUT_NC    0
#define OUT_H     (BSZ*VOCAB)                // 16384
#define OUT_ALPHA (OUT_H + 2*BSZ*HID)        // 81920

// ---------------------------------------------------------------------------
// Kernel 1: per-batch scalar bias  sbias[b] = prev_h.w_h + y.w_y + align_b,
// and gather y[b] = emb[prev_y[b]].
__global__ void bias_y_kernel(const float* __restrict__ prev_s,
                              const int*   __restrict__ prev_y,
                              const float* __restrict__ emb,
                              const float* __restrict__ align_w,
                              const float* __restrict__ align_b,
                              float* __restrict__ y_out,
                              float* __restrict__ sbias) {
  const int b = blockIdx.x;
  const int t = threadIdx.x;                 // 256 threads
  __shared__ float red[256];
  const int yi = prev_y[b];
  if (t < EMB) y_out[b*EMB + t] = emb[yi*EMB + t];
  float acc = 0.0f;
  // prev_h = [prev_s[0][b][:] , prev_s[1][b][:]]  (2*HID = 1024)
  for (int i = t; i < 2*HID; i += 256) {
    const int dir = i >> 9, j = i & (HID-1);
    acc += prev_s[(dir*BSZ + b)*HID + j] * align_w[i];
  }
  for (int i = t; i < EMB; i += 256)
    acc += emb[yi*EMB + i] * align_w[2*HID + i];
  red[t] = acc; __syncthreads();
  for (int s = 128; s > 0; s >>= 1) {
    if (t < s) red[t] += red[t + s];
    __syncthreads();
  }
  if (t == 0) sbias[b] = red[0] + align_b[0];
}

// ---------------------------------------------------------------------------
// Kernel 2: scores[b,s] = sbias[b] + dot(z[b,s,:], w_z)   (z streaming pass 1)
// One wave per (b,s) row; lane-parallel float4 loads over CTX=512.
__global__ void scores_kernel(const float* __restrict__ z,
                              const float* __restrict__ align_w,
                              const float* __restrict__ sbias,
                              float* __restrict__ scores) {
  const int gwave = (int)((blockIdx.x * blockDim.x + threadIdx.x) >> 5);
  const int lane  = threadIdx.x & 31;
  const int b = gwave >> 12;          // SEQ = 4096 rows per batch
  const int s = gwave & (SEQ-1);
  const float* zp = z + ((size_t)b*SEQ + s)*CTX;
  const float* wz = align_w + 2*HID + EMB;   // +1152
  float acc = 0.0f;
#pragma unroll
  for (int i = 0; i < 4; ++i) {
    const float4 zv = *(const float4*)(zp + (i*32 + lane)*4);
    const float4 wv = *(const float4*)(wz + (i*32 + lane)*4);
    acc += zv.x*wv.x + zv.y*wv.y + zv.z*wv.z + zv.w*wv.w;
  }
#pragma unroll
  for (int off = 16; off > 0; off >>= 1)
    acc += __shfl_xor(acc, off, 32);
  if (lane == 0) scores[gwave] = acc + sbias[b];
}

// ---------------------------------------------------------------------------
// Kernel 3: softmax over S per batch; alpha written straight into d_out.
__global__ void softmax_kernel(const float* __restrict__ scores,
                               float* __restrict__ alpha) {
  const int b = blockIdx.x;
  const int t = threadIdx.x;                 // 256 threads
  __shared__ float red[256];
  const float* sp = scores + b*SEQ;
  float m = -INFINITY;
  for (int i = t; i < SEQ; i += 256) m = fmaxf(m, sp[i]);
  red[t] = m; __syncthreads();
  for (int s = 128; s > 0; s >>= 1) {
    if (t < s) red[t] = fmaxf(red[t], red[t + s]);
    __syncthreads();
  }
  m = red[0]; __syncthreads();
  float sum = 0.0f;
  for (int i = t; i < SEQ; i += 256) {
    const float e = __expf(sp[i] - m);
    alpha[b*SEQ + i] = e;
    sum += e;
  }
  red[t] = sum; __syncthreads();
  for (int s = 128; s > 0; s >>= 1) {
    if (t < s) red[t] += red[t + s];
    __syncthreads();
  }
  const float inv = 1.0f / red[0];
  for (int i = t; i < SEQ; i += 256) alpha[b*SEQ + i] *= inv;
}

// ---------------------------------------------------------------------------
// Kernel 4: partial context (z streaming pass 2).
// grid = (B, 8 s-parts), 128 threads; each thread owns 4 consecutive d
// columns (float4 / b128 loads). alpha chunks staged in LDS via the CDNA5
// async-load path (GLOBAL_LOAD_ASYNC_TO_LDS_B32, ASYNCcnt).
__global__ void context_part_kernel(const float* __restrict__ z,
                                    const float* __restrict__ alpha,
                                    float* __restrict__ cpart) {
  const int b    = blockIdx.x;
  const int part = blockIdx.y;               // 0..7
  const int t    = threadIdx.x;              // 128 threads
  __shared__ float as[128];
  const float* zp = z + (size_t)b*SEQ*CTX + t*4;
  const int s_beg = part * (SEQ/8);
  float4 acc = make_float4(0.0f, 0.0f, 0.0f, 0.0f);
  lds_f32* lds_dst = (lds_f32*)&as[t];
  for (int s0 = s_beg; s0 < s_beg + SEQ/8; s0 += 128) {
    const float* ap = alpha + b*SEQ + s0 + t;
    asm volatile("global_load_async_to_lds_b32 %0, %1, off"
                 :: "v"(lds_dst), "v"(ap) : "memory");
    asm volatile("s_wait_asynccnt 0x0" ::: "memory");
    __syncthreads();
#pragma unroll 4
    for (int s = 0; s < 128; ++s) {
      const float  a  = as[s];
      const float4 zv = *(const float4*)(zp + (size_t)(s0 + s)*CTX);
      acc.x = fmaf(a, zv.x, acc.x);
      acc.y = fmaf(a, zv.y, acc.y);
      acc.z = fmaf(a, zv.z, acc.z);
      acc.w = fmaf(a, zv.w, acc.w);
    }
    __syncthreads();
  }
  *(float4*)(cpart + (size_t)(b*8 + part)*CTX + t*4) = acc;
}

__global__ void context_reduce_kernel(const float* __restrict__ cpart,
                                      float* __restrict__ context) {
  const int e = blockIdx.x * blockDim.x + threadIdx.x;   // B*CTX
  const int b = e / CTX, d = e % CTX;
  float acc = 0.0f;
#pragma unroll
  for (int p = 0; p < 8; ++p) acc += cpart[(size_t)(b*8 + p)*CTX + d];
  context[e] = acc;
}

// ---------------------------------------------------------------------------
// Generic fp32 WMMA GEMM:  C[M x N] = A[M x K] @ W[N x K]^T (+ bias[N])
// One 16x16 tile per wave, K stepped by 4 via V_WMMA_F32_16X16X4_F32.
__global__ void gemm_wmma_kernel(const float* __restrict__ A,
                                 const float* __restrict__ W,
                                 const float* __restrict__ bias,
                                 float* __restrict__ C,
                                 int M, int N, int K) {
  const int wave = (int)((blockIdx.x * blockDim.x + threadIdx.x) >> 5);
  const int lane = threadIdx.x & 31;
  const int ntn  = N >> 4;
  const int mt   = wave / ntn;
  const int nt   = wave - mt * ntn;
  if (mt * 16 >= M) return;                   // wave-uniform exit
  const int m0 = mt << 4, n0 = nt << 4;
  const int half = lane >> 4;                 // 0|1
  const int l16  = lane & 15;
  const float* arow = A + (size_t)(m0 + l16) * K + half*2;
  const float* brow = W + (size_t)(n0 + l16) * K + half*2;
  v8f acc = {};
  for (int k0 = 0; k0 < K; k0 += 4) {
    const v2f a = *(const v2f*)(arow + k0);   // A[m0+l16][k0+half*2 .. +1]
    const v2f b = *(const v2f*)(brow + k0);   // B[k][n] = W[n0+l16][k0+half*2 ..]
    acc = __builtin_amdgcn_wmma_f32_16x16x4_f32(
        /*neg_a=*/false, a, /*neg_b=*/false, b,
        /*c_mod=*/(short)0, acc, /*reuse_a=*/false, /*reuse_b=*/false);
  }
  const int n = n0 + l16;
  const float bv = bias ? bias[n] : 0.0f;
#pragma unroll
  for (int v = 0; v < 8; ++v) {
    const int m = m0 + v + half*8;            // C/D VGPR layout
    C[(size_t)m * N + n] = acc[v] + bv;
  }
}

// ---------------------------------------------------------------------------
// Kernel: GRU elementwise combine.
// h'[dir,b,i] from gx, gh (64x1536 each per dir), biases, prev_s.
__global__ void gru_elem_kernel(const float* __restrict__ gx,
                                const float* __restrict__ gh,
                                const float* __restrict__ bih,
                                const float* __restrict__ bhh,
                                const float* __restrict__ prev_s,
                                float* __restrict__ h_out) {
  const int e   = blockIdx.x * blockDim.x + threadIdx.x;   // 2*B*HID
  const int dir = e >> 15;                                 // /(B*HID)
  const int rem = e & (BSZ*HID - 1);
  const int b   = rem >> 9;
  const int i   = rem & (HID-1);
  const size_t gbase = (size_t)(dir*BSZ + b) * G3;
  const size_t bbase = (size_t)dir * G3;
  const float rsum = gx[gbase + i]         + bih[bbase + i]
                   + gh[gbase + i]         + bhh[bbase + i];
  const float zsum = gx[gbase + HID + i]   + bih[bbase + HID + i]
                   + gh[gbase + HID + i]   + bhh[bbase + HID + i];
  const float nx   = gx[gbase + 2*HID + i] + bih[bbase + 2*HID + i];
  const float nh   = gh[gbase + 2*HID + i] + bhh[bbase + 2*HID + i];
  const float r  = 1.0f / (1.0f + __expf(-rsum));
  const float zg = 1.0f / (1.0f + __expf(-zsum));
  const float n  = tanhf(nx + r * nh);
  const float hp = prev_s[(dir*BSZ + b)*HID + i];
  h_out[e] = (1.0f - zg) * n + zg * hp;
}

// ---------------------------------------------------------------------------
extern "C" void kernel_launch(void* const* d_in, const int* in_sizes, int n_in,
                              void* d_out, int out_size, void* d_ws, size_t ws_size,
                              hipStream_t stream) {
  const float* prev_s  = (const float*)d_in[0];
  const int*   prev_y  = (const int*)  d_in[1];
  const float* z       = (const float*)d_in[2];
  const float* emb     = (const float*)d_in[3];
  const float* align_w = (const float*)d_in[4];
  const float* align_b = (const float*)d_in[5];
  const float* gru_wih = (const float*)d_in[6];
  const float* gru_whh = (const float*)d_in[7];
  const float* gru_bih = (const float*)d_in[8];
  const float* gru_bhh = (const float*)d_in[9];
  const float* w1      = (const float*)d_in[10];
  const float* b1      = (const float*)d_in[11];
  const float* w2      = (const float*)d_in[12];
  const float* b2      = (const float*)d_in[13];

  float* ws  = (float*)d_ws;
  float* out = (float*)d_out;

  float* scores  = ws + OFF_SCORES;
  float* sbias   = ws + OFF_SBIAS;
  float* yv      = ws + OFF_Y;
  float* cpart   = ws + OFF_CPART;
  float* context = ws + OFF_CTX;
  float* gx      = ws + OFF_GX;
  float* gh      = ws + OFF_GH;
  float* tmid    = ws + OFF_T;

  float* nc_out    = out + OUT_NC;
  float* h_out     = out + OUT_H;
  float* alpha_out = out + OUT_ALPHA;

  // 1. per-batch bias + embedding gather
  bias_y_kernel<<<BSZ, 256, 0, stream>>>(prev_s, prev_y, emb, align_w, align_b,
                                         yv, sbias);

  // 2. scores (z pass 1): one wave per (b,s); 8 waves/block
  scores_kernel<<<(BSZ*SEQ)/8, 256, 0, stream>>>(z, align_w, sbias, scores);

  // 3. softmax -> alpha (directly in d_out)
  softmax_kernel<<<BSZ, 256, 0, stream>>>(scores, alpha_out);

  // 4. context (z pass 2): float4 columns, async-LDS alpha staging, 8 S-parts
  context_part_kernel<<<dim3(BSZ, 8), 128, 0, stream>>>(z, alpha_out, cpart);
  context_reduce_kernel<<<(BSZ*CTX)/256, 256, 0, stream>>>(cpart, context);

  // 5. GRU GEMMs (fp32 WMMA): gx = y @ wih^T, gh = prev_s @ whh^T, per dir
  {
    const int tiles_g = (BSZ/16) * (G3/16);          // 4*96 = 384
    const int blocks_g = (tiles_g + 7) / 8;          // 8 waves/block
    for (int dir = 0; dir < 2; ++dir) {
      gemm_wmma_kernel<<<blocks_g, 256, 0, stream>>>(
          yv, gru_wih + (size_t)dir*G3*EMB, nullptr,
          gx + (size_t)dir*BSZ*G3, BSZ, G3, EMB);
      gemm_wmma_kernel<<<blocks_g, 256, 0, stream>>>(
          prev_s + (size_t)dir*BSZ*HID, gru_whh + (size_t)dir*G3*HID, nullptr,
          gh + (size_t)dir*BSZ*G3, BSZ, G3, HID);
    }
  }

  // 6. GRU elementwise -> h (d_out)
  gru_elem_kernel<<<(2*BSZ*HID)/256, 256, 0, stream>>>(gx, gh, gru_bih, gru_bhh,
                                                       prev_s, h_out);

  // 7. MLP head (fp32 WMMA)
  {
    const int tiles1 = (BSZ/16) * (D1/16);           // 256
    gemm_wmma_kernel<<<(tiles1 + 7)/8, 256, 0, stream>>>(
        context, w1, b1, tmid, BSZ, D1, CTX);
    const int tiles2 = (BSZ/16) * (VOCAB/16);        // 64
    gemm_wmma_kernel<<<(tiles2 + 7)/8, 256, 0, stream>>>(
        tmid, w2, b2, nc_out, BSZ, VOCAB, D1);
  }

  (void)in_sizes; (void)n_in; (void)out_size; (void)ws_size;
}